// KoopmanOperator_56203942035611
// MI455X (gfx1250) — compile-verified
//
#include <hip/hip_runtime.h>
#include <hip/hip_bf16.h>

#define DT 0.01f

typedef __attribute__((ext_vector_type(2))) float v2f;
typedef __attribute__((ext_vector_type(8))) float v8f;

constexpr int BATCH    = 16384;
constexpr int TSTEPS   = 64;
constexpr int DDIM     = 32;
constexpr int WAVES    = 4;            // waves per block
constexpr int ROWS_W   = 16;           // batch rows per wave (one WMMA M-tile)
constexpr int RSTRIDE  = 36;           // padded LDS row stride in words (bank-conflict-free)

// One 16x16 f32 output tile of (A[16x32] @ W[32x32])[:, n*16:(n+1)*16] via 8 chained K=4 WMMAs.
// Weight fragments were pre-staged in LDS as [mat][n][k][lane] float2.
__device__ inline v8f gemm_tile(const v2f ya[8], const float* ldsW, int mat, int n,
                                int lane, float bias) {
    v8f acc = {bias, bias, bias, bias, bias, bias, bias, bias};
    const float* base = ldsW + (size_t)((mat * 2 + n) * 8) * 64 + lane * 2;
#pragma unroll
    for (int k = 0; k < 8; ++k) {
        v2f wb = *(const v2f*)(base + k * 64);
        acc = __builtin_amdgcn_wmma_f32_16x16x4_f32(
            /*neg_a=*/false, ya[k], /*neg_b=*/false, wb,
            /*c_mod=*/(short)0, acc, /*reuse_a=*/false, /*reuse_b=*/false);
    }
    return acc;
}

__global__ __launch_bounds__(WAVES * 32)
void koopman_kernel(const float* __restrict__ x,
                    const float* __restrict__ cw1, const float* __restrict__ cb1,
                    const float* __restrict__ cw2, const float* __restrict__ cb2,
                    const float* __restrict__ rw1, const float* __restrict__ rb1,
                    const float* __restrict__ rw2, const float* __restrict__ rb2,
                    float* __restrict__ out) {
    // Weight fragments: 4 mats x 2 n-tiles x 8 k-frags x 32 lanes x float2 = 16 KB
    __shared__ float ldsW[4 * 2 * 8 * 64];
    // Per-wave staging for C-layout -> A-layout transpose (padded rows)
    __shared__ float ldsS[WAVES * 16 * RSTRIDE];

    const int tid  = threadIdx.x;
    const int lane = tid & 31;
    const int wave = tid >> 5;
    const int l16  = lane & 15;
    const int hi   = lane >> 4;      // 0: K pair {0,1} / M 0..7 ; 1: K pair {2,3} / M 8..15

    // ---- Stage the four 32x32 weight matrices into B-fragment layout (once) ----
    const float* Ws[4] = {cw1, cw2, rw1, rw2};
    {
        const float* W = Ws[wave];   // WAVES==4: one matrix per wave
#pragma unroll
        for (int n = 0; n < 2; ++n)
#pragma unroll
            for (int k = 0; k < 8; ++k) {
                int col  = n * 16 + l16;
                int krow = k * 4 + 2 * hi;
                int idx  = ((wave * 2 + n) * 8 + k) * 64 + lane * 2;
                ldsW[idx + 0] = W[krow * 32 + col];
                ldsW[idx + 1] = W[(krow + 1) * 32 + col];
            }
    }
    __syncthreads();

    // Biases depend only on N -> one scalar per n-tile per lane
    float cb1v[2], cb2v[2], rb1v[2], rb2v[2];
#pragma unroll
    for (int n = 0; n < 2; ++n) {
        cb1v[n] = cb1[n * 16 + l16];
        cb2v[n] = cb2[n * 16 + l16];
        rb1v[n] = rb1[n * 16 + l16];
        rb2v[n] = rb2[n * 16 + l16];
    }

    const int rowBase = (blockIdx.x * WAVES + wave) * ROWS_W;
    float* stage = &ldsS[wave * 16 * RSTRIDE];
    const bool odd = (lane & 1);

    // ---- Load initial state y = x[:, 0, :] in WMMA C-layout ----
    v8f yc[2];
#pragma unroll
    for (int n = 0; n < 2; ++n)
#pragma unroll
        for (int r = 0; r < 8; ++r) {
            int M = r + 8 * hi;
            yc[n][r] = x[(size_t)(rowBase + M) * TSTEPS * DDIM + (n * 16 + l16)];
        }

#pragma unroll 1
    for (int t = 0; t < TSTEPS; ++t) {
        // ---- C-layout y -> LDS ----
#pragma unroll
        for (int n = 0; n < 2; ++n)
#pragma unroll
            for (int r = 0; r < 8; ++r)
                stage[(r + 8 * hi) * RSTRIDE + n * 16 + l16] = yc[n][r];

        // ---- A-fragments of y (ds_load_b64 each) ----
        v2f ya[8];
#pragma unroll
        for (int k = 0; k < 8; ++k)
            ya[k] = *(const v2f*)&stage[l16 * RSTRIDE + k * 4 + 2 * hi];

        // ---- complex-path MLP: h = relu(y@cw1+cb1); p = h@cw2+cb2 ----
        v8f hc0 = gemm_tile(ya, ldsW, 0, 0, lane, cb1v[0]);
        v8f hc1 = gemm_tile(ya, ldsW, 0, 1, lane, cb1v[1]);
#pragma unroll
        for (int r = 0; r < 8; ++r) { hc0[r] = fmaxf(hc0[r], 0.f); hc1[r] = fmaxf(hc1[r], 0.f); }
#pragma unroll
        for (int r = 0; r < 8; ++r) {
            stage[(r + 8 * hi) * RSTRIDE + l16]      = hc0[r];
            stage[(r + 8 * hi) * RSTRIDE + 16 + l16] = hc1[r];
        }
        v2f ha[8];
#pragma unroll
        for (int k = 0; k < 8; ++k)
            ha[k] = *(const v2f*)&stage[l16 * RSTRIDE + k * 4 + 2 * hi];
        v8f p[2];
        p[0] = gemm_tile(ha, ldsW, 1, 0, lane, cb2v[0]);
        p[1] = gemm_tile(ha, ldsW, 1, 1, lane, cb2v[1]);

        // ---- real-path MLP: re = relu(y@rw1+rb1)@rw2+rb2 (reuses ya) ----
        v8f hr0 = gemm_tile(ya, ldsW, 2, 0, lane, rb1v[0]);
        v8f hr1 = gemm_tile(ya, ldsW, 2, 1, lane, rb1v[1]);
#pragma unroll
        for (int r = 0; r < 8; ++r) { hr0[r] = fmaxf(hr0[r], 0.f); hr1[r] = fmaxf(hr1[r], 0.f); }
#pragma unroll
        for (int r = 0; r < 8; ++r) {
            stage[(r + 8 * hi) * RSTRIDE + l16]      = hr0[r];
            stage[(r + 8 * hi) * RSTRIDE + 16 + l16] = hr1[r];
        }
#pragma unroll
        for (int k = 0; k < 8; ++k)
            ha[k] = *(const v2f*)&stage[l16 * RSTRIDE + k * 4 + 2 * hi];
        v8f re[2];
        re[0] = gemm_tile(ha, ldsW, 3, 0, lane, rb2v[0]);
        re[1] = gemm_tile(ha, ldsW, 3, 1, lane, rb2v[1]);

        // ---- elementwise update; pair (2j, 2j+1) lives in adjacent lanes ----
#pragma unroll
        for (int n = 0; n < 2; ++n)
#pragma unroll
            for (int r = 0; r < 8; ++r) {
                float pv = p[n][r];
                float pn = __shfl_xor(pv, 1, 32);
                float mu = odd ? pn : pv;
                float om = odd ? pv : pn;
                float e  = __expf(DT * mu);
                float c  = __cosf(DT * om) * e;
                float s  = __sinf(DT * om) * e;
                float yv = yc[n][r];
                float yn = __shfl_xor(yv, 1, 32);
                float y0 = odd ? yn : yv;
                float y1 = odd ? yv : yn;
                float comp = odd ? (s * y0 + c * y1) : (c * y0 - s * y1);
                float real = __expf(DT * re[n][r]) * yv;
                float ynew = 0.5f * (comp + real);
                yc[n][r] = ynew;
                int M = r + 8 * hi;
                out[(size_t)(rowBase + M) * TSTEPS * DDIM + (size_t)t * DDIM + n * 16 + l16] = ynew;
            }
    }
}

extern "C" void kernel_launch(void* const* d_in, const int* in_sizes, int n_in,
                              void* d_out, int out_size, void* d_ws, size_t ws_size,
                              hipStream_t stream) {
    // setup_inputs order: x, T, cw1, cb1, cw2, cb2, rw1, rb1, rw2, rb2
    const float* x   = (const float*)d_in[0];
    const float* cw1 = (const float*)d_in[2];
    const float* cb1 = (const float*)d_in[3];
    const float* cw2 = (const float*)d_in[4];
    const float* cb2 = (const float*)d_in[5];
    const float* rw1 = (const float*)d_in[6];
    const float* rb1 = (const float*)d_in[7];
    const float* rw2 = (const float*)d_in[8];
    const float* rb2 = (const float*)d_in[9];
    float* out = (float*)d_out;

    dim3 grid(BATCH / (WAVES * ROWS_W));   // 256 blocks
    dim3 block(WAVES * 32);                // 128 threads = 4 waves
    koopman_kernel<<<grid, block, 0, stream>>>(x, cw1, cb1, cw2, cb2,
                                               rw1, rb1, rw2, rb2, out);
}